// Sparsemax_58282706206943
// MI455X (gfx1250) — compile-verified
//
#include <hip/hip_runtime.h>
#include <hip/hip_bf16.h>
#include <stdint.h>

// Sparsemax over rows of length 1024, f32.
// Algorithm: Michelot fixed-point (exact, no sort):
//   tau0 = (sum(z) - 1)/d
//   repeat: S = {z > tau}; tau = (sum_S z - 1)/|S|; until |S| stable
//   p = max(z - tau, 0)
//
// Mapping: 1 wave32 per row; lane holds 32 elements in VGPRs.
// Data path: global -> LDS via CDNA5 async copy engine (ASYNCcnt),
//            LDS -> VGPR via ds_load_b128, VGPR -> global via b128 stores.

#define D               1024
#define ROWS_PER_BLOCK  8          // 8 wave32s per 256-thread block
#define ELEMS_PER_LANE  32         // D / 32
#define CHUNKS          8          // ELEMS_PER_LANE / 4 (float4 chunks)

__device__ __forceinline__ void wait_asynccnt0() {
#if __has_builtin(__builtin_amdgcn_s_wait_asynccnt)
    __builtin_amdgcn_s_wait_asynccnt(0);
#else
    asm volatile("s_wait_asynccnt 0" ::: "memory");
#endif
}

__global__ __launch_bounds__(256)
void sparsemax_rows_kernel(const float* __restrict__ z,
                           float* __restrict__ p,
                           int nrows) {
    __shared__ float lds[ROWS_PER_BLOCK * D];   // 32 KB per block

    const int wave = threadIdx.x >> 5;          // 0..7
    const int lane = threadIdx.x & 31;          // wave32 lane
    const int row  = blockIdx.x * ROWS_PER_BLOCK + wave;
    if (row >= nrows) return;                   // wave-uniform exit

    const float* __restrict__ zrow = z + (size_t)row * D;
    float*       __restrict__ prow = p + (size_t)row * D;
    float* lrow = &lds[wave * D];

    // ---- Async DMA: global row -> this wave's private LDS slice ----------
    // Each issue: 32 lanes x 16 B = 512 B; 8 issues cover the 4 KB row.
    // Low 32 bits of a flat pointer into LDS == workgroup-relative LDS addr.
    {
        const uint32_t lbase = (uint32_t)(uintptr_t)lrow;
        const uint64_t gbase = (uint64_t)(uintptr_t)zrow;
        const uint32_t lane16 = (uint32_t)(lane << 4);
#pragma unroll
        for (int j = 0; j < CHUNKS; ++j) {
            uint32_t laddr = lbase + (uint32_t)(j * 512) + lane16;
            uint64_t gaddr = gbase + (uint64_t)(j * 512 + (lane << 4));
            asm volatile("global_load_async_to_lds_b128 %0, %1, off"
                         :: "v"(laddr), "v"(gaddr)
                         : "memory");
        }
        wait_asynccnt0();   // per-wave ASYNCcnt; no workgroup barrier needed
    }

    // ---- LDS -> registers: lane owns 8 float4 chunks (32 elements) -------
    float v[ELEMS_PER_LANE];
#pragma unroll
    for (int j = 0; j < CHUNKS; ++j) {
        const float4 q = *reinterpret_cast<const float4*>(&lrow[j * 128 + lane * 4]);
        v[j * 4 + 0] = q.x;
        v[j * 4 + 1] = q.y;
        v[j * 4 + 2] = q.z;
        v[j * 4 + 3] = q.w;
    }

    // ---- Initial tau from full-row sum -----------------------------------
    float s = 0.0f;
#pragma unroll
    for (int e = 0; e < ELEMS_PER_LANE; ++e) s += v[e];
#pragma unroll
    for (int off = 16; off > 0; off >>= 1) s += __shfl_xor(s, off, 32);

    float tau   = (s - 1.0f) * (1.0f / (float)D);
    int prev_cnt = D;

    // ---- Michelot fixed-point iterations (wave-uniform control flow) -----
    for (int iter = 0; iter < 64; ++iter) {
        float ps = 0.0f, pc = 0.0f;
#pragma unroll
        for (int e = 0; e < ELEMS_PER_LANE; ++e) {
            const bool m = v[e] > tau;
            ps += m ? v[e] : 0.0f;
            pc += m ? 1.0f : 0.0f;
        }
#pragma unroll
        for (int off = 16; off > 0; off >>= 1) {
            ps += __shfl_xor(ps, off, 32);
            pc += __shfl_xor(pc, off, 32);
        }
        const int   cnt     = (int)pc;            // exact: cnt <= 1024
        const float tau_new = (ps - 1.0f) / pc;
        tau = tau_new;
        if (cnt == prev_cnt) break;               // support stable -> exact tau
        prev_cnt = cnt;
    }

    // ---- p = max(z - tau, 0), coalesced b128 stores ----------------------
#pragma unroll
    for (int j = 0; j < CHUNKS; ++j) {
        float4 q;
        q.x = fmaxf(v[j * 4 + 0] - tau, 0.0f);
        q.y = fmaxf(v[j * 4 + 1] - tau, 0.0f);
        q.z = fmaxf(v[j * 4 + 2] - tau, 0.0f);
        q.w = fmaxf(v[j * 4 + 3] - tau, 0.0f);
        *reinterpret_cast<float4*>(&prow[j * 128 + lane * 4]) = q;
    }
}

extern "C" void kernel_launch(void* const* d_in, const int* in_sizes, int n_in,
                              void* d_out, int out_size, void* d_ws, size_t ws_size,
                              hipStream_t stream) {
    (void)n_in; (void)d_ws; (void)ws_size; (void)out_size;
    const float* zin = (const float*)d_in[0];
    float* out = (float*)d_out;
    const int nrows = in_sizes[0] / D;                       // 32768
    const int grid  = (nrows + ROWS_PER_BLOCK - 1) / ROWS_PER_BLOCK;
    sparsemax_rows_kernel<<<grid, 256, 0, stream>>>(zin, out, nrows);
}